// DiagonalTraining_41197326303254
// MI455X (gfx1250) — compile-verified
//
#include <hip/hip_runtime.h>

// ---------------------------------------------------------------------------
// DiagonalTraining on MI455X (gfx1250, wave32)
//
// For each diagonal i: out[b,k] = sum_j D[b,j] * W[i,k,j] + bias[i,k],
// D[b,j] = x[b, j, i-j] (j<=i), scatter out[b,k] -> x[b, i-k, k].
//
// Memory-bound: W triangular stream ~190MB @ 23.3TB/s -> ~9us floor.
// W chunks streamed global->LDS via Tensor Data Mover (double-buffered,
// HW pad 2 dwords / 32 dwords -> bank-conflict-free row stride 34),
// exact f32 math via V_WMMA_F32_16X16X4_F32 (M padded 8->16).
// ---------------------------------------------------------------------------

#define SDIM 512
#define BATCH 8
#define KC 32                 // j-columns of W per TDM chunk
#define DSTRIDE 516           // 516%64==4 -> conflict-free A-frag column reads
#define WSTRIDE (KC + 2)      // 34 (TDM pad): 8B-aligned, conflict-free B-frags

typedef __attribute__((ext_vector_type(2))) float v2f;
typedef __attribute__((ext_vector_type(8))) float v8f;
typedef __attribute__((ext_vector_type(4))) unsigned int v4u;
typedef __attribute__((ext_vector_type(4))) int v4i;
typedef __attribute__((ext_vector_type(8))) int v8i;

// Issue one TDM 2D tile load: 128 rows x KC cols of f32, row stride SDIM,
// LDS row stride WSTRIDE dwords (pad_enable: +2 dwords every 32 dwords).
__device__ __forceinline__ void tdm_issue_w(unsigned lds_addr, const float* gptr)
{
    const unsigned long long ga = (unsigned long long)(uintptr_t)gptr;
    v4u g0;
    g0[0] = 1u;                                  // count=1, user mode
    g0[1] = lds_addr;                            // LDS dest byte address
    g0[2] = (unsigned)ga;                        // global_addr[31:0]
    g0[3] = (unsigned)((ga >> 32) & 0x01FFFFFFu) // global_addr[56:32]
          | (2u << 30);                          // type=2 ("image")
    v8i g1;
    g1[0] = (int)((2u << 16)                     // data_size = 4B
                | (1u << 20)                     // pad_enable
                | (4u << 22)                     // pad_interval: 32 dwords
                | (1u << 25));                   // pad_amount: 2 dwords
    g1[1] = (int)((unsigned)KC << 16);           // tensor_dim0 = KC
    g1[2] = (int)(128u << 16);                   // tensor_dim1 = 128
    g1[3] = (int)((unsigned)KC << 16);           // tile_dim0 = KC
    g1[4] = 128;                                 // tile_dim1 = 128 (tile_dim2=0)
    g1[5] = SDIM;                                // tensor_dim0_stride[31:0]
    g1[6] = 0;
    g1[7] = 0;
    const v4i z4 = {0, 0, 0, 0};                 // groups 2/3 unused (2D)
    const v8i z8 = {0, 0, 0, 0, 0, 0, 0, 0};
    __builtin_amdgcn_tensor_load_to_lds(g0, g1, z4, z4, z8, 0);
}

__global__ __launch_bounds__(256)
void diag_linear_wmma_kernel(const float* __restrict__ x,
                             const float* __restrict__ W,
                             const float* __restrict__ bias,
                             float* __restrict__ out)
{
    const int i  = blockIdx.y;        // diagonal index 0..511
    const int qx = blockIdx.x;        // k-quadrant: k in [qx*128, qx*128+128)
    if (qx * 128 > i) return;         // uniform: skip zero triangle of W

    __shared__ float Ds[9 * DSTRIDE];             // 8 gathered rows + 1 zero row
    __shared__ float Wl[2][128 * WSTRIDE];        // TDM double buffer

    const int tid  = threadIdx.x;
    const int lane = tid & 31;
    const int wave = tid >> 5;

    const size_t wbase = (size_t)i * SDIM * SDIM + (size_t)(qx * 128) * SDIM;
    const int nchunks = i / KC + 1;

    // Kick off chunk 0 via TDM while all threads gather D.
    if (wave == 0) tdm_issue_w((unsigned)(uintptr_t)&Wl[0][0], &W[wbase]);

    // Stage D: Ds[b][j] = x[b, j, i-j] for b<8 & j<=i, else 0; row 8 = zeros
    // (rows 8..15 of the A tile alias row 8 -> broadcast reads, no divergence).
    for (int e = tid; e < 9 * SDIM; e += 256) {
        const int b = e >> 9;
        const int j = e & (SDIM - 1);
        float v = 0.0f;
        if (b < BATCH && j <= i)
            v = x[(size_t)b * SDIM * SDIM + (size_t)j * SDIM + (i - j)];
        Ds[b * DSTRIDE + j] = v;
    }

    // Wave w owns out columns [k0, k0+16)
    const int  k0     = qx * 128 + wave * 16;
    const bool active = (k0 <= i);

    const int m     = lane & 15;        // A/C row (batch), B col (N)
    const int khalf = (lane >> 4) * 2;  // lanes 0-15 -> K{0,1}, 16-31 -> K{2,3}
    const int arow  = (m < BATCH) ? m : BATCH;   // zero row for pad batches

    // Accumulator initialized with bias (broadcast over batch rows)
    float bv = 0.0f;
    if (active) bv = bias[(size_t)i * SDIM + k0 + m];
    v8f c;
    c[0] = bv; c[1] = bv; c[2] = bv; c[3] = bv;
    c[4] = bv; c[5] = bv; c[6] = bv; c[7] = bv;

    for (int t = 0; t < nchunks; ++t) {
        const int jc  = t * KC;
        const int cur = t & 1;

        __syncthreads();   // prev compute done with buf[cur^1]; D staged (t==0)
        if (wave == 0) {
            if (t + 1 < nchunks) {
                tdm_issue_w((unsigned)(uintptr_t)&Wl[cur ^ 1][0],
                            &W[wbase + (size_t)(jc + KC)]);
                __builtin_amdgcn_s_wait_tensorcnt(1);   // chunk t complete
            } else {
                __builtin_amdgcn_s_wait_tensorcnt(0);
            }
        }
        __syncthreads();   // buf[cur] ready for all waves

        if (active) {
            const float* wrow = &Wl[cur][(wave * 16 + m) * WSTRIDE];
            const float* drow = &Ds[arow * DSTRIDE + jc];
            #pragma unroll
            for (int j0 = 0; j0 < KC; j0 += 4) {
                // A fragment (16x4 f32): (M=m, K=jc+j0+khalf, +1)
                v2f a;
                a[0] = drow[j0 + khalf];
                a[1] = drow[j0 + khalf + 1];
                // B fragment (4x16 f32): B[K=j, N=n] = W[i, k0+n, jc+j0+j]
                v2f bf;
                bf[0] = wrow[j0 + khalf];
                bf[1] = wrow[j0 + khalf + 1];
                c = __builtin_amdgcn_wmma_f32_16x16x4_f32(
                        false, a, false, bf, (short)0, c, false, false);
            }
        }
    }

    // Scatter: out[b, i-k, k] = c[M=b, N=k-k0]; C layout: VGPR v = row M=v
    // in lanes 0-15 (N=lane), rows 8-15 (padding) in lanes 16-31, dropped.
    if (active && lane < 16) {
        const int k = k0 + m;
        if (k <= i) {
            const int r = i - k;
            #pragma unroll
            for (int b = 0; b < BATCH; ++b)
                out[(size_t)b * SDIM * SDIM + (size_t)r * SDIM + k] = c[b];
        }
    }
}

// out = x for untouched positions (r+c >= S); diag kernel overwrites the rest.
__global__ __launch_bounds__(256)
void copy_x_kernel(const float* __restrict__ x, float* __restrict__ out, int n4)
{
    const int idx = blockIdx.x * 256 + threadIdx.x;
    if (idx < n4) ((float4*)out)[idx] = ((const float4*)x)[idx];
}

extern "C" void kernel_launch(void* const* d_in, const int* in_sizes, int n_in,
                              void* d_out, int out_size, void* d_ws, size_t ws_size,
                              hipStream_t stream)
{
    const float* x    = (const float*)d_in[0];
    const float* W    = (const float*)d_in[1];
    const float* bias = (const float*)d_in[2];
    float* out = (float*)d_out;

    const int n4 = BATCH * SDIM * SDIM / 4;   // 8 MB passthrough copy
    copy_x_kernel<<<(n4 + 255) / 256, 256, 0, stream>>>(x, out, n4);

    dim3 grid(SDIM / 128, SDIM);              // (4 k-quadrants, 512 diagonals)
    diag_linear_wmma_kernel<<<grid, 256, 0, stream>>>(x, W, bias, out);
}